// SKA_20950850470021
// MI455X (gfx1250) — compile-verified
//
#include <hip/hip_runtime.h>
#include <cstdint>

// Involution / spatial-kernel-attention:
//   out[b, i*G+g, h, w] = sum_{dy,dx} x[b, i*G+g, h+dy-1, w+dx-1] * wgt[b, g, dy*3+dx, h, w]
// B=8, C=256, G=32 (cpg=8), H=W=56, K=3. Memory-bound (~80 MB moved, ~116 MFLOP).
// Strategy: stage x (with row halo) and w tiles in LDS via CDNA5 async
// global->LDS b128 copies, reuse out of LDS (x: 9x, w: 8x), compute fp32.

#define B_      8
#define C_      256
#define G_      32
#define CPG     8          // channels per group = C/G
#define H_      56
#define W_      56
#define K2_     9
#define TH      8          // output rows per block
#define NT      (H_ / TH)  // 7 row-tiles (exact)
#define THREADS 256
#define ROWQ    (W_ / 4)   // 14 float4 per row (56 floats, 224 B, 16B-aligned rows)

__global__ __launch_bounds__(THREADS)
void SKA_involution_kernel(const float* __restrict__ x,
                           const float* __restrict__ wgt,
                           float* __restrict__ out)
{
    // xs: rows h0-1 .. h0+TH (10 rows) per channel; no column padding
    // (column halo handled at compute time so every LDS b128 store stays aligned)
    __shared__ float xs[CPG][TH + 2][W_];   // 17920 B
    __shared__ float ws[K2_][TH][W_];       // 16128 B

    const int tid = threadIdx.x;
    int blk = blockIdx.x;
    const int ht = blk % NT;  blk /= NT;
    const int g  = blk % G_;  blk /= G_;
    const int b  = blk;                   // 0..B_-1
    const int h0 = ht * TH;

    // ---------------- Stage tiles into LDS ----------------
#if defined(__gfx1250__)
    // CDNA5 async global->LDS path (ASYNCcnt-tracked, bypasses VGPRs).
    // x tile: CPG * 10 rows * 14 float4 = 1120 b128 transfers
    {
        const int NXJ = CPG * (TH + 2) * ROWQ;
        for (int j = tid; j < NXJ; j += THREADS) {
            const int i   = j / ((TH + 2) * ROWQ);
            const int rem = j % ((TH + 2) * ROWQ);
            const int rl  = rem / ROWQ;          // local row 0..9
            const int q   = rem % ROWQ;          // float4 within row
            const int gh  = h0 - 1 + rl;         // global row (may be OOB)
            if ((unsigned)gh < (unsigned)H_) {
                const float* gp =
                    x + ((((size_t)b * C_ + (size_t)i * G_ + g) * H_ + gh) * W_ + q * 4);
                const uint32_t loff = (uint32_t)(uintptr_t)&xs[i][rl][q * 4];
                asm volatile("global_load_async_to_lds_b128 %0, %1, off"
                             :: "v"(loff), "v"(gp) : "memory");
            }
        }
    }
    // w tile: 9 taps * 8 rows * 14 float4 = 1008 b128 transfers
    {
        const int NWJ = K2_ * TH * ROWQ;
        for (int j = tid; j < NWJ; j += THREADS) {
            const int k   = j / (TH * ROWQ);
            const int rem = j % (TH * ROWQ);
            const int rl  = rem / ROWQ;
            const int q   = rem % ROWQ;
            const float* gp =
                wgt + ((((size_t)b * G_ + g) * K2_ + k) * H_ + (h0 + rl)) * W_ + q * 4;
            const uint32_t loff = (uint32_t)(uintptr_t)&ws[k][rl][q * 4];
            asm volatile("global_load_async_to_lds_b128 %0, %1, off"
                         :: "v"(loff), "v"(gp) : "memory");
        }
    }
    // Zero the (at most one) out-of-image halo row. Plain ds stores; these
    // target exactly the LDS rows the asyncs skipped, so no ordering hazard.
    if (h0 == 0) {
        for (int j = tid; j < CPG * ROWQ; j += THREADS) {
            const float4 z = {0.f, 0.f, 0.f, 0.f};
            *(float4*)&xs[j / ROWQ][0][(j % ROWQ) * 4] = z;
        }
    }
    if (h0 + TH >= H_) {
        for (int j = tid; j < CPG * ROWQ; j += THREADS) {
            const float4 z = {0.f, 0.f, 0.f, 0.f};
            *(float4*)&xs[j / ROWQ][TH + 1][(j % ROWQ) * 4] = z;
        }
    }
    // Wave-local async completion; barrier below makes it workgroup-wide.
    asm volatile("s_wait_asynccnt 0" ::: "memory");
#else
    // Fallback (host parse / non-gfx1250): synchronous staging.
    {
        const int NXJ = CPG * (TH + 2) * ROWQ;
        for (int j = tid; j < NXJ; j += THREADS) {
            const int i   = j / ((TH + 2) * ROWQ);
            const int rem = j % ((TH + 2) * ROWQ);
            const int rl  = rem / ROWQ;
            const int q   = rem % ROWQ;
            const int gh  = h0 - 1 + rl;
            float4 v = {0.f, 0.f, 0.f, 0.f};
            if ((unsigned)gh < (unsigned)H_) {
                v = *(const float4*)(x + ((((size_t)b * C_ + (size_t)i * G_ + g) * H_ + gh) * W_ + q * 4));
            }
            *(float4*)&xs[i][rl][q * 4] = v;
        }
        const int NWJ = K2_ * TH * ROWQ;
        for (int j = tid; j < NWJ; j += THREADS) {
            const int k   = j / (TH * ROWQ);
            const int rem = j % (TH * ROWQ);
            const int rl  = rem / ROWQ;
            const int q   = rem % ROWQ;
            *(float4*)&ws[k][rl][q * 4] =
                *(const float4*)(wgt + ((((size_t)b * G_ + g) * K2_ + k) * H_ + (h0 + rl)) * W_ + q * 4);
        }
    }
#endif
    __syncthreads();

    // ---------------- Compute: 8 ch * 8 rows * 56 cols = 3584 outputs ----------------
    for (int idx = tid; idx < CPG * TH * W_; idx += THREADS) {
        const int col = idx % W_;
        const int r   = (idx / W_) % TH;
        const int i   = idx / (W_ * TH);
        float acc = 0.f;
#pragma unroll
        for (int dy = 0; dy < 3; ++dy) {
            const float* xrow = xs[i][r + dy];   // buf row r+dy == global row h0+r+dy-1
#pragma unroll
            for (int dx = 0; dx < 3; ++dx) {
                const int c2 = col + dx - 1;
                const float xv = ((unsigned)c2 < (unsigned)W_) ? xrow[c2] : 0.f;
                acc = fmaf(xv, ws[dy * 3 + dx][r][col], acc);
            }
        }
        out[(((size_t)b * C_ + (size_t)i * G_ + g) * H_ + (h0 + r)) * W_ + col] = acc;
    }
}

extern "C" void kernel_launch(void* const* d_in, const int* in_sizes, int n_in,
                              void* d_out, int out_size, void* d_ws, size_t ws_size,
                              hipStream_t stream) {
    (void)in_sizes; (void)n_in; (void)out_size; (void)d_ws; (void)ws_size;
    const float* x   = (const float*)d_in[0];
    const float* wgt = (const float*)d_in[1];
    float* out       = (float*)d_out;

    const dim3 grid(B_ * G_ * NT);   // 8*32*7 = 1792 blocks
    const dim3 block(THREADS);       // 8 wave32s
    hipLaunchKernelGGL(SKA_involution_kernel, grid, block, 0, stream, x, wgt, out);
}